// YoloLoss_79018808312101
// MI455X (gfx1250) — compile-verified
//
#include <hip/hip_runtime.h>
#include <math.h>

// ---------------- problem constants ----------------
#define NTOT   10647            // 507 + 2028 + 8112
#define GT     20
#define BQ     32
#define NCHUNK 16
#define CHUNK  672              // 16 * 672 = 10752 >= 10647, multiple of 16

typedef float v2f __attribute__((ext_vector_type(2)));
typedef float v8f __attribute__((ext_vector_type(8)));

// floor(anchor / sf) * sf, precomputed per [scale*3 + anchor]
__constant__ float c_AW[9] = {96.f,128.f,352.f, 16.f,48.f,48.f, 8.f,16.f,32.f};
__constant__ float c_AH[9] = {64.f,192.f,320.f, 48.f,32.f,112.f, 8.f,24.f,16.f};

// map global anchor index n -> source pointer (channel 0), grid pos, scale consts.
// Divisions are done per-branch against literal constants so clang emits
// mul-hi/shift instead of a generic divide.
__device__ __forceinline__ void locate(int n, int b,
                                       const float* __restrict__ x0,
                                       const float* __restrict__ x1,
                                       const float* __restrict__ x2,
                                       const float** src, int* WWo,
                                       float* gxo, float* gyo, float* awsf, float* ahsf)
{
    int s, a, WW; size_t pos; float gx, gy; const float* base;
    if (n < 507) {
        int ns = n;            WW = 169;  base = x0;
        a = ns / 169;          int yx = ns - a * 169;
        int y = yx / 13;       int x  = yx - y * 13;
        gx = (float)x; gy = (float)y; s = 0; pos = (size_t)yx;
        *src = base + ((size_t)b * 255 + (size_t)a * 85) * 169 + pos;
    } else if (n < 2535) {
        int ns = n - 507;      WW = 676;  base = x1;
        a = ns / 676;          int yx = ns - a * 676;
        int y = yx / 26;       int x  = yx - y * 26;
        gx = (float)x; gy = (float)y; s = 1; pos = (size_t)yx;
        *src = base + ((size_t)b * 255 + (size_t)a * 85) * 676 + pos;
    } else {
        int ns = n - 2535;     WW = 2704; base = x2;
        a = ns / 2704;         int yx = ns - a * 2704;
        int y = yx / 52;       int x  = yx - y * 52;
        gx = (float)x; gy = (float)y; s = 2; pos = (size_t)yx;
        *src = base + ((size_t)b * 255 + (size_t)a * 85) * 2704 + pos;
    }
    *WWo  = WW;
    *gxo  = gx; *gyo = gy;
    *awsf = c_AW[s * 3 + a];
    *ahsf = c_AH[s * 3 + a];
}

__device__ __forceinline__ float sigm(float v) { return 1.f / (1.f + expf(-v)); }

// scale factor per n (needed only in decode)
__device__ __forceinline__ float scale_of(int n) {
    return (n < 507) ? 32.f : ((n < 2535) ? 16.f : 8.f);
}

// ---------------- stage 1: decode + IoU/argmax + WMMA noobj accumulation ----------------
__global__ __launch_bounds__(256)
void yolo_stage1(const float* __restrict__ x0, const float* __restrict__ x1,
                 const float* __restrict__ x2, const float* __restrict__ by,
                 float4* __restrict__ partial)
{
    __shared__ float4 sBox[CHUNK];
    __shared__ float  sLog[CHUNK];
    __shared__ float  sNo[8][16];
    __shared__ float  sCnt[8][16];
    __shared__ float  sMax[8][16];
    __shared__ int    sIdx[8][16];

    const int wg     = blockIdx.x;            // b * NCHUNK + chunk
    const int b      = wg / NCHUNK;
    const int chunk  = wg - b * NCHUNK;
    const int cstart = chunk * CHUNK;
    const int t      = threadIdx.x;

    // ---- cooperative decode of this chunk into LDS (only channels 0..4 touched) ----
    for (int i = t; i < CHUNK; i += 256) {
        int n = cstart + i;
        float4 box = make_float4(0.f, 0.f, 0.f, 0.f);
        float  lg  = 0.f;
        if (n < NTOT) {
            const float* src; int WW; float gx, gy, awsf, ahsf;
            locate(n, b, x0, x1, x2, &src, &WW, &gx, &gy, &awsf, &ahsf);
            float sf = scale_of(n);
            float tx = src[0], ty = src[WW], tw = src[2 * WW], th = src[3 * WW], tc = src[4 * WW];
            box.x = (sigm(tx) + gx) * sf;
            box.y = (sigm(ty) + gy) * sf;
            box.z = expf(tw) * awsf;
            box.w = expf(th) * ahsf;
            float conf = sigm(tc);
            lg = logf(1.f - conf + 1e-9f);
        }
        sBox[i] = box;
        sLog[i] = lg;
    }
    __syncthreads();

    // ---- WMMA loop: A(16x4)=neg mask, B(4x16) col0=log1m col1=1 -> D[g][0]=noobj, D[g][1]=cnt
    const int wave  = t >> 5;
    const int lane  = t & 31;
    const int half  = lane >> 4;      // selects K pair {0,1} vs {2,3}
    const int row   = lane & 15;      // A row (g) and B column id
    const int gbase = (wave & 1) * 16;
    const int g     = gbase + row;
    const bool gval = (g < GT);
    const int gg    = gval ? g : 0;
    const float* gt = by + ((size_t)b * GT + gg) * 5;
    const float gx = gt[0], gy = gt[1], gw = gt[2], gh = gt[3];
    const float garea = gw * gh;
    const float gxr = gx + gw, gyb = gy + gh;

    // hoisted B-column selectors: col0 takes log1m, col1 takes 1.0, others 0
    const bool  isCol0   = (row == 0);
    const float colConst = (row == 1) ? 1.f : 0.f;

    const int pair = wave >> 1;                 // 4 wave-pairs split the chunk
    const int sub0 = pair * (CHUNK / 4);
    const int sub1 = sub0 + (CHUNK / 4);

    v8f acc = {};
    float mv = -INFINITY;
    int   mi = 0x7fffffff;

#pragma unroll 2
    for (int i0 = sub0; i0 < sub1; i0 += 4) {
        const int i = i0 + 2 * half;            // this lane evaluates n = cstart+i, cstart+i+1
        float4 p0 = sBox[i];
        float4 p1 = sBox[i + 1];
        // unconditional 8B LDS load for all lanes (i is even -> aligned), no EXEC games
        float2 lg2 = *(const float2*)&sLog[i];

        float xl = fmaxf(p0.x, gx), yt = fmaxf(p0.y, gy);
        float xr = fminf(p0.x + p0.z, gxr), ybb = fminf(p0.y + p0.w, gyb);
        float in0 = fmaxf(xr - xl, 0.f) * fmaxf(ybb - yt, 0.f);
        float iou0 = in0 * __builtin_amdgcn_rcpf(p0.z * p0.w + garea - in0);

        xl = fmaxf(p1.x, gx); yt = fmaxf(p1.y, gy);
        xr = fminf(p1.x + p1.z, gxr); ybb = fminf(p1.y + p1.w, gyb);
        float in1 = fmaxf(xr - xl, 0.f) * fmaxf(ybb - yt, 0.f);
        float iou1 = in1 * __builtin_amdgcn_rcpf(p1.z * p1.w + garea - in1);

        const int n0 = cstart + i, n1 = n0 + 1;
        const bool v0 = (n0 < NTOT), v1 = (n1 < NTOT);

        float a0 = ((int)gval & (int)v0 & (int)(iou0 < 0.5f)) ? 1.f : 0.f;
        float a1 = ((int)gval & (int)v1 & (int)(iou1 < 0.5f)) ? 1.f : 0.f;
        float b0 = isCol0 ? lg2.x : colConst;
        float b1 = isCol0 ? lg2.y : colConst;

        v2f Av = {a0, a1};
        v2f Bv = {b0, b1};
        acc = __builtin_amdgcn_wmma_f32_16x16x4_f32(false, Av, false, Bv,
                                                    (short)0, acc, false, false);

        // branchless first-occurrence argmax (v_cmp + v_cndmask, no EXEC games)
        bool bet0 = ((int)gval & (int)v0 &
                     ((int)(iou0 > mv) | ((int)(iou0 == mv) & (int)(n0 < mi)))) != 0;
        mv = bet0 ? iou0 : mv;
        mi = bet0 ? n0   : mi;
        bool bet1 = ((int)gval & (int)v1 &
                     ((int)(iou1 > mv) | ((int)(iou1 == mv) & (int)(n1 < mi)))) != 0;
        mv = bet1 ? iou1 : mv;
        mi = bet1 ? n1   : mi;
    }

    // combine argmax with the partner lane (same g, other K pair)
    {
        float omv = __shfl_xor(mv, 16, 32);
        int   omi = __shfl_xor(mi, 16, 32);
        bool tk = ((int)(omv > mv) | ((int)(omv == mv) & (int)(omi < mi))) != 0;
        mv = tk ? omv : mv;
        mi = tk ? omi : mi;
    }
    if (lane < 16) { sMax[wave][row] = mv; sIdx[wave][row] = mi; }

    // extract D columns 0 (noobj) and 1 (cnt): lanes {0,16} and {1,17}
    if (row == 0) { for (int r = 0; r < 8; ++r) sNo[wave][half * 8 + r]  = acc[r]; }
    if (row == 1) { for (int r = 0; r < 8; ++r) sCnt[wave][half * 8 + r] = acc[r]; }
    __syncthreads();

    // cross-wave combine, write per-(b,chunk,g) partial
    if (t < GT) {
        const int wbase = (t < 16) ? 0 : 1;
        const int r     = (t < 16) ? t : (t - 16);
        float no = 0.f, cn = 0.f, bmv = -INFINITY;
        int bmi = 0x7fffffff;
        for (int w = wbase; w < 8; w += 2) {
            no += sNo[w][r];
            cn += sCnt[w][r];
            float wm = sMax[w][r]; int wi = sIdx[w][r];
            bool tk = ((int)(wm > bmv) | ((int)(wm == bmv) & (int)(wi < bmi))) != 0;
            bmv = tk ? wm : bmv;
            bmi = tk ? wi : bmi;
        }
        partial[(size_t)wg * GT + t] = make_float4(no, cn, bmv, __int_as_float(bmi));
    }
}

// ---------------- stage 2: combine chunks, gather at argmax, finalize loss ----------------
__global__ __launch_bounds__(256)
void yolo_stage2(const float* __restrict__ x0, const float* __restrict__ x1,
                 const float* __restrict__ x2, const float* __restrict__ by,
                 const float4* __restrict__ partial, float* __restrict__ out)
{
    int tid = blockIdx.x * blockDim.x + threadIdx.x;
    if (tid >= BQ * GT) return;
    const int b = tid / GT;
    const int g = tid - b * GT;

    float no = 0.f, cn = 0.f, mv = -INFINITY;
    int mi = 0x7fffffff;
    for (int c = 0; c < NCHUNK; ++c) {
        float4 p = partial[((size_t)(b * NCHUNK + c)) * GT + g];
        no += p.x; cn += p.y;
        int pi = __float_as_int(p.w);
        bool tk = ((int)(p.z > mv) | ((int)(p.z == mv) & (int)(pi < mi))) != 0;
        mv = tk ? p.z : mv;
        mi = tk ? pi  : mi;
    }
    const float noobj = no / fmaxf(cn, 1.f);

    const float* src; int WW; float gxp, gyp, awsf, ahsf;
    locate(mi, b, x0, x1, x2, &src, &WW, &gxp, &gyp, &awsf, &ahsf);

    const float* gt = by + ((size_t)b * GT + g) * 5;
    float dx = src[0]      - gt[0];
    float dy = src[WW]     - gt[1];
    float dw = src[2 * WW] - gt[2];
    float dh = src[3 * WW] - gt[3];
    const float mse = 0.25f * (dx * dx + dy * dy + dw * dw + dh * dh);

    const float conf = sigm(src[4 * WW]);
    const float obj  = logf(conf + 1e-9f);

    int lab = (int)gt[4];
    lab = lab < 0 ? 0 : (lab > 79 ? 79 : lab);
    float m = -INFINITY, vlab = 0.f;
    for (int c = 0; c < 80; ++c) {
        float v = sigm(src[(5 + c) * WW]);
        m = fmaxf(m, v);
        vlab = (c == lab) ? v : vlab;
    }
    float se = 0.f;
    for (int c = 0; c < 80; ++c) {
        float v = sigm(src[(5 + c) * WW]);
        se += expf(v - m);
    }
    const float ce = -(vlab - (m + logf(se)));

    atomicAdd(out, mse - obj - noobj - ce);
}

__global__ void yolo_zero(float* out) { out[0] = 0.f; }

// ---------------- host launch ----------------
extern "C" void kernel_launch(void* const* d_in, const int* in_sizes, int n_in,
                              void* d_out, int out_size, void* d_ws, size_t ws_size,
                              hipStream_t stream)
{
    const float* x0 = (const float*)d_in[0];
    const float* x1 = (const float*)d_in[1];
    const float* x2 = (const float*)d_in[2];
    const float* by = (const float*)d_in[3];
    float* out      = (float*)d_out;
    float4* partial = (float4*)d_ws;   // BQ*NCHUNK*GT float4 = 160 KB

    yolo_zero<<<1, 1, 0, stream>>>(out);
    yolo_stage1<<<BQ * NCHUNK, 256, 0, stream>>>(x0, x1, x2, by, partial);
    yolo_stage2<<<(BQ * GT + 255) / 256, 256, 0, stream>>>(x0, x1, x2, by, partial, out);
}